// GraphConvolutionCVAE_61306363183556
// MI455X (gfx1250) — compile-verified
//
#include <hip/hip_runtime.h>
#include <hip/hip_bf16.h>
#include <math.h>

// ---------------- problem constants ----------------
#define N_NODES 50000
#define N_EDGES 400000
#define NF 2
#define PS 3
#define HID 512
#define LAT 64
#define KK 5           // GMM kernel size
#define EPS_G 1e-15f
#define COND_PAD 68    // LAT+PS=67 padded to multiple of 4

typedef __attribute__((ext_vector_type(2))) float v2f;
typedef __attribute__((ext_vector_type(8))) float v8f;

// ---------------- utility ----------------
__global__ void zero_kernel(float* __restrict__ p, int n) {
  int i = blockIdx.x * blockDim.x + threadIdx.x;
  if (i < n) p[i] = 0.0f;
}

// copy src[0..n_src) into dst[0..n_dst), zero-filling the padded tail
__global__ void pad_copy_kernel(const float* __restrict__ src, float* __restrict__ dst,
                                int n_src, int n_dst) {
  int i = blockIdx.x * blockDim.x + threadIdx.x;
  if (i < n_dst) dst[i] = (i < n_src) ? src[i] : 0.0f;
}

// normalize node feats, material params, edge pseudo-coords in one pass
__global__ void normalize_kernel(const float* __restrict__ x, const float* __restrict__ mat,
                                 const float* __restrict__ ef,
                                 const float* __restrict__ nm, const float* __restrict__ ns,
                                 const float* __restrict__ pm, const float* __restrict__ pstd,
                                 const float* __restrict__ em, const float* __restrict__ es,
                                 float* __restrict__ nnf, float* __restrict__ nmat,
                                 float* __restrict__ nedge) {
  int i = blockIdx.x * blockDim.x + threadIdx.x;
  if (i < N_NODES * NF) { int c = i % NF; nnf[i]  = (x[i]   - nm[c]) / ns[c]; }
  if (i < N_NODES * PS) { int c = i % PS; nmat[i] = (mat[i] - pm[c]) / pstd[c]; }
  if (i < N_EDGES)      {                 nedge[i] = (ef[i] - em[0]) / es[0]; }
}

__global__ void deg_kernel(const int* __restrict__ dst, float* __restrict__ deg) {
  int e = blockIdx.x * blockDim.x + threadIdx.x;
  if (e < N_EDGES) atomicAdd(&deg[dst[e]], 1.0f);
}

__global__ void concat_kernel(const float* __restrict__ nnf, const float* __restrict__ nmat,
                              float* __restrict__ h5) {
  int n = blockIdx.x * blockDim.x + threadIdx.x;
  if (n < N_NODES) {
    h5[n * 5 + 0] = nnf[n * 2 + 0];
    h5[n * 5 + 1] = nnf[n * 2 + 1];
    h5[n * 5 + 2] = nmat[n * 3 + 0];
    h5[n * 5 + 3] = nmat[n * 3 + 1];
    h5[n * 5 + 4] = nmat[n * 3 + 2];
  }
}

// ---------------- GMM conv pieces ----------------
// xt[n, k*NF + c] = sum_i xin[n,i] * g[i, k*NF + c]
__global__ void conv_xt_kernel(const float* __restrict__ xin, const float* __restrict__ g,
                               float* __restrict__ xt, int cin) {
  int n = blockIdx.x * blockDim.x + threadIdx.x;
  if (n >= N_NODES) return;
  float acc[KK * NF];
#pragma unroll
  for (int kc = 0; kc < KK * NF; ++kc) acc[kc] = 0.0f;
  for (int i = 0; i < cin; ++i) {
    float v = xin[n * cin + i];
#pragma unroll
    for (int kc = 0; kc < KK * NF; ++kc) acc[kc] += v * g[i * (KK * NF) + kc];
  }
#pragma unroll
  for (int kc = 0; kc < KK * NF; ++kc) xt[n * (KK * NF) + kc] = acc[kc];
}

// per-edge: gaussian weights on the fly, gather xt[src], f32 atomic scatter into agg[dst]
__global__ void conv_edge_kernel(const float* __restrict__ xt, const int* __restrict__ src,
                                 const int* __restrict__ dst, const float* __restrict__ nedge,
                                 const float* __restrict__ mu, const float* __restrict__ sigma,
                                 float* __restrict__ agg) {
  int e = blockIdx.x * blockDim.x + threadIdx.x;
  if (e >= N_EDGES) return;
  int s = src[e], d = dst[e];
  float p = nedge[e];
  const float* xs = xt + (size_t)s * (KK * NF);
  float m0 = 0.0f, m1 = 0.0f;
#pragma unroll
  for (int k = 0; k < KK; ++k) {
    float diff = p - mu[k];
    float sg = sigma[k];
    float w = expf(-0.5f * diff * diff / (EPS_G + sg * sg));
    m0 += xs[k * NF + 0] * w;
    m1 += xs[k * NF + 1] * w;
  }
  atomicAdd(&agg[d * 2 + 0], m0);
  atomicAdd(&agg[d * 2 + 1], m1);
}

// out = [elu](agg/deg + xin@root + bias) + res
__global__ void conv_finish_kernel(const float* __restrict__ agg, const float* __restrict__ deg,
                                   const float* __restrict__ xin, const float* __restrict__ root,
                                   const float* __restrict__ bias, const float* __restrict__ res,
                                   float* __restrict__ out, int cin, int do_elu) {
  int n = blockIdx.x * blockDim.x + threadIdx.x;
  if (n >= N_NODES) return;
  float dg = deg[n]; if (dg < 1.0f) dg = 1.0f;
  float o0 = bias[0], o1 = bias[1];
  for (int i = 0; i < cin; ++i) {
    float v = xin[n * cin + i];
    o0 += v * root[i * 2 + 0];
    o1 += v * root[i * 2 + 1];
  }
  o0 += agg[n * 2 + 0] / dg;
  o1 += agg[n * 2 + 1] / dg;
  if (do_elu) {
    o0 = (o0 > 0.0f) ? o0 : expm1f(o0);
    o1 = (o1 > 0.0f) ? o1 : expm1f(o1);
  }
  out[n * 2 + 0] = o0 + res[n * 2 + 0];
  out[n * 2 + 1] = o1 + res[n * 2 + 1];
}

// ---------------- big GEMVs (bandwidth path, streaming FMA + prefetch) ----------------
// ypre[512] += x[r0:r0+256] @ W[r0:r0+256, :512]   (205 MB stream, HBM-bound)
__global__ void gemv_big_kernel(const float* __restrict__ x, const float* __restrict__ W,
                                float* __restrict__ ypre, int din) {
  __shared__ float xs[256];
  int t = threadIdx.x;
  int r0 = blockIdx.x * 256;
  int nr = din - r0; if (nr > 256) nr = 256;
  xs[t] = (t < nr) ? x[r0 + t] : 0.0f;
  __syncthreads();
  float a0 = 0.0f, a1 = 0.0f;
  const float* Wp = W + (size_t)r0 * HID + t;
  for (int r = 0; r < nr; ++r) {
    __builtin_prefetch(Wp + (size_t)HID * 24, 0, 1);   // global_prefetch_b8 ahead of stream
    float v = xs[r];
    a0 += v * Wp[0];
    a1 += v * Wp[256];
    Wp += HID;
  }
  atomicAdd(&ypre[t], a0);
  atomicAdd(&ypre[t + 256], a1);
}

__global__ void bias_act_kernel(const float* __restrict__ ypre, const float* __restrict__ b,
                                float* __restrict__ y, int n, int relu) {
  int i = blockIdx.x * blockDim.x + threadIdx.x;
  if (i < n) {
    float v = ypre[i] + b[i];
    if (relu && v < 0.0f) v = 0.0f;
    y[i] = v;
  }
}

// out[j] = b[j] + sum_i h[i] * W[i, j],  W is [HID, dout], coalesced along j
__global__ void gemv_wide_kernel(const float* __restrict__ h, const float* __restrict__ W,
                                 const float* __restrict__ b, float* __restrict__ out, int dout) {
  __shared__ float hs[HID];
  int t = threadIdx.x;
  hs[t] = h[t];
  hs[t + 256] = h[t + 256];
  __syncthreads();
  int j = blockIdx.x * 256 + t;
  if (j >= dout) return;
  float acc = b[j];
  const float* Wp = W + j;
  for (int i = 0; i < HID; ++i) {
    __builtin_prefetch(Wp + (size_t)dout * 16, 0, 1);
    acc += hs[i] * Wp[0];
    Wp += dout;
  }
  out[j] = acc;
}

// ---------------- small dense layers on the CDNA5 matrix pipe ----------------
// GEMV mapped onto V_WMMA_F32_16X16X4_F32 (full f32 precision, wave32).
// REQUIRES din % 4 == 0 (call sites pad). Branch-free inner loop:
//   2 uniform s_load for broadcast activations, arithmetic lane-masking
//   (sel0/sel16 are loop-invariant VGPRs, so each mul/fma uses <=1 SGPR),
//   2 unconditional global_load_b32 for the 4x16 B tile, 1 v_wmma per step.
//   A 16x4: row M=0 only -> lane0 carries {x[k],x[k+1]}, lane16 {x[k+2],x[k+3]}
//   B 4x16: VGPR0 = rows K0 (lanes 0-15) / K2 (lanes 16-31), VGPR1 = K1 / K3
//   D 16x16: row M=0 is c[0] on lanes 0-15 -> 16 output columns per wave
__global__ void wmma_gemv_kernel(const float* __restrict__ x, const float* __restrict__ W,
                                 const float* __restrict__ b, float* __restrict__ out,
                                 int din, int dout, int relu) {
  int lane = threadIdx.x;                 // 0..31, one wave per block
  int col0 = blockIdx.x * 16;
  int col  = col0 + (lane & 15);
  int koff = (lane < 16) ? 0 : 2;
  float sel0  = (lane == 0)  ? 1.0f : 0.0f;   // loop-invariant lane masks (VGPRs)
  float sel16 = (lane == 16) ? 1.0f : 0.0f;
  v8f c = {};
  const float* Wp = W + (size_t)koff * dout + col;
  for (int k = 0; k < din; k += 4) {
    float x0 = x[k + 0];
    float x1 = x[k + 1];
    float x2 = x[k + 2];
    float x3 = x[k + 3];
    v2f a;
    a.x = sel0 * x0 + sel16 * x2;
    a.y = sel0 * x1 + sel16 * x3;
    v2f bb;
    bb.x = Wp[0];
    bb.y = Wp[(size_t)dout];
    Wp += (size_t)4 * dout;
    c = __builtin_amdgcn_wmma_f32_16x16x4_f32(false, a, false, bb, (short)0, c, false, false);
  }
  if (lane < 16) {
    float v = c[0] + b[col];
    if (relu && v < 0.0f) v = 0.0f;
    out[col] = v;
  }
}

// layernorm over 512 features (gamma/beta), in place; single block of 256
__global__ void layernorm_kernel(float* __restrict__ h, const float* __restrict__ gamma,
                                 const float* __restrict__ beta) {
  __shared__ float red[256];
  int t = threadIdx.x;
  float v0 = h[t], v1 = h[t + 256];
  red[t] = v0 + v1;
  __syncthreads();
  for (int s = 128; s > 0; s >>= 1) { if (t < s) red[t] += red[t + s]; __syncthreads(); }
  float m = red[0] / 512.0f;
  __syncthreads();
  red[t] = (v0 - m) * (v0 - m) + (v1 - m) * (v1 - m);
  __syncthreads();
  for (int s = 128; s > 0; s >>= 1) { if (t < s) red[t] += red[t + s]; __syncthreads(); }
  float var = red[0] / 512.0f;
  float rs = rsqrtf(var + 1e-5f);
  h[t]       = (v0 - m) * rs * gamma[t]       + beta[t];
  h[t + 256] = (v1 - m) * rs * gamma[t + 256] + beta[t + 256];
}

// z = mu + eps * exp(0.5*logvar); cond = [z, nmat[0,:3], 0-pad to COND_PAD]
__global__ void reparam_kernel(const float* __restrict__ mu, const float* __restrict__ lv,
                               const float* __restrict__ eps, const float* __restrict__ nmat,
                               float* __restrict__ cond) {
  int t = threadIdx.x;
  if (t < LAT) cond[t] = mu[t] + eps[t] * expf(0.5f * lv[t]);
  else if (t < LAT + PS) cond[t] = nmat[t - LAT];
  else if (t < COND_PAD) cond[t] = 0.0f;
}

// ---------------- host launch ----------------
extern "C" void kernel_launch(void* const* d_in, const int* in_sizes, int n_in,
                              void* d_out, int out_size, void* d_ws, size_t ws_size,
                              hipStream_t stream) {
  (void)in_sizes; (void)n_in; (void)out_size; (void)ws_size;

  // ---- input leaves (jax pytree flattening of setup_inputs dict) ----
  const float* x    = (const float*)d_in[0];
  const float* mat  = (const float*)d_in[1];
  const float* ef   = (const float*)d_in[2];
  const int*   eidx = (const int*)d_in[3];
  const int*   src  = eidx;
  const int*   dstp = eidx + N_EDGES;
  // d_in[4] = is_training (unused: reference always reparameterizes with eps)
  const float* nm   = (const float*)d_in[5];
  const float* ns   = (const float*)d_in[6];
  const float* pm   = (const float*)d_in[7];
  const float* pstd = (const float*)d_in[8];
  const float* em   = (const float*)d_in[9];
  const float* es   = (const float*)d_in[10];
  // conv_params[c] = (g, mu, sigma, root, bias) at 11 + 5*c
  const float* cg[4], *cmu[4], *csg[4], *crt[4], *cbi[4];
  for (int c = 0; c < 4; ++c) {
    cg[c]  = (const float*)d_in[11 + 5 * c + 0];
    cmu[c] = (const float*)d_in[11 + 5 * c + 1];
    csg[c] = (const float*)d_in[11 + 5 * c + 2];
    crt[c] = (const float*)d_in[11 + 5 * c + 3];
    cbi[c] = (const float*)d_in[11 + 5 * c + 4];
  }
  // mlp blocks: (W1,b1,W2,b2,gamma,beta,W3,b3)
  const float* MU[8]; const float* LV[8]; const float* UP[8];
  for (int j = 0; j < 8; ++j) {
    MU[j] = (const float*)d_in[31 + j];
    LV[j] = (const float*)d_in[39 + j];
    UP[j] = (const float*)d_in[47 + j];
  }
  const float* epsv = (const float*)d_in[55];

  // ---- workspace layout (floats) ----
  float* ws = (float*)d_ws;
  float* nnf   = ws + 0;        // 100000
  float* nmat  = ws + 100000;   // 150000
  float* nedge = ws + 250000;   // 400000
  float* deg   = ws + 650000;   // 50000
  float* h5    = ws + 700000;   // 250000
  float* xt    = ws + 950000;   // 500000
  float* agg   = ws + 1450000;  // 100000
  float* h1    = ws + 1550000;  // 100000
  float* h2    = ws + 1650000;  // 100000  (== flat)
  float* dvec  = ws + 1750000;  // 100000  (decoded)
  float* ybuf  = ws + 1850000;  // 100000
  float* ypre  = ws + 1950000;  // 512 (reused mu then lv)
  float* y1    = ws + 1950512;  // 512
  float* y2mu  = ws + 1951024;  // 512
  float* muv   = ws + 1951536;  // 64
  float* y2lv  = ws + 1951600;  // 512
  float* lvv   = ws + 1952112;  // 64
  float* cond  = ws + 1952176;  // 80 (COND_PAD used)
  float* u1    = ws + 1952256;  // 512
  float* u2    = ws + 1952768;  // 512
  float* w1pad = ws + 1953280;  // COND_PAD*HID = 34816 (zero-padded upW1)

  const int gN   = (N_NODES + 255) / 256;        // 196
  const int gE   = (N_EDGES + 255) / 256;        // 1563
  const int gBig = (N_NODES * NF + 255) / 256;   // 391
  const int gPad = (COND_PAD * HID + 255) / 256; // 136

  // ---- normalize + degree + concat + pad upW1 ----
  zero_kernel<<<gN, 256, 0, stream>>>(deg, N_NODES);
  pad_copy_kernel<<<gPad, 256, 0, stream>>>(UP[0], w1pad, (LAT + PS) * HID, COND_PAD * HID);
  normalize_kernel<<<gE, 256, 0, stream>>>(x, mat, ef, nm, ns, pm, pstd, em, es,
                                           nnf, nmat, nedge);
  deg_kernel<<<gE, 256, 0, stream>>>(dstp, deg);
  concat_kernel<<<gN, 256, 0, stream>>>(nnf, nmat, h5);

  // ---- encoder conv 0: [N,5] -> [N,2], elu + nnf ----
  conv_xt_kernel<<<gN, 256, 0, stream>>>(h5, cg[0], xt, NF + PS);
  zero_kernel<<<gBig, 256, 0, stream>>>(agg, N_NODES * NF);
  conv_edge_kernel<<<gE, 256, 0, stream>>>(xt, src, dstp, nedge, cmu[0], csg[0], agg);
  conv_finish_kernel<<<gN, 256, 0, stream>>>(agg, deg, h5, crt[0], cbi[0], nnf, h1, NF + PS, 1);

  // ---- encoder conv 1: [N,2] -> [N,2], elu + nnf ----
  conv_xt_kernel<<<gN, 256, 0, stream>>>(h1, cg[1], xt, NF);
  zero_kernel<<<gBig, 256, 0, stream>>>(agg, N_NODES * NF);
  conv_edge_kernel<<<gE, 256, 0, stream>>>(xt, src, dstp, nedge, cmu[1], csg[1], agg);
  conv_finish_kernel<<<gN, 256, 0, stream>>>(agg, deg, h1, crt[1], cbi[1], nnf, h2, NF, 1);

  // ---- mu MLP: flat[100000] -> 512 -> 512 -> LN -> 64 ----
  zero_kernel<<<2, 256, 0, stream>>>(ypre, HID);
  gemv_big_kernel<<<gBig, 256, 0, stream>>>(h2, MU[0], ypre, N_NODES * NF);
  bias_act_kernel<<<2, 256, 0, stream>>>(ypre, MU[1], y1, HID, 1);
  wmma_gemv_kernel<<<HID / 16, 32, 0, stream>>>(y1, MU[2], MU[3], y2mu, HID, HID, 1);
  layernorm_kernel<<<1, 256, 0, stream>>>(y2mu, MU[4], MU[5]);
  wmma_gemv_kernel<<<LAT / 16, 32, 0, stream>>>(y2mu, MU[6], MU[7], muv, HID, LAT, 0);

  // ---- logvar MLP ----
  zero_kernel<<<2, 256, 0, stream>>>(ypre, HID);
  gemv_big_kernel<<<gBig, 256, 0, stream>>>(h2, LV[0], ypre, N_NODES * NF);
  bias_act_kernel<<<2, 256, 0, stream>>>(ypre, LV[1], y1, HID, 1);
  wmma_gemv_kernel<<<HID / 16, 32, 0, stream>>>(y1, LV[2], LV[3], y2lv, HID, HID, 1);
  layernorm_kernel<<<1, 256, 0, stream>>>(y2lv, LV[4], LV[5]);
  wmma_gemv_kernel<<<LAT / 16, 32, 0, stream>>>(y2lv, LV[6], LV[7], lvv, HID, LAT, 0);

  // ---- reparameterize + decode latent: [68 padded] -> 512 -> 512 -> LN -> [N*NF] ----
  reparam_kernel<<<1, 96, 0, stream>>>(muv, lvv, epsv, nmat, cond);
  wmma_gemv_kernel<<<HID / 16, 32, 0, stream>>>(cond, w1pad, UP[1], u1, COND_PAD, HID, 1);
  wmma_gemv_kernel<<<HID / 16, 32, 0, stream>>>(u1, UP[2], UP[3], u2, HID, HID, 1);
  layernorm_kernel<<<1, 256, 0, stream>>>(u2, UP[4], UP[5]);
  gemv_wide_kernel<<<gBig, 256, 0, stream>>>(u2, UP[6], UP[7], dvec, N_NODES * NF);

  // ---- decoder conv 2: elu + d ----
  conv_xt_kernel<<<gN, 256, 0, stream>>>(dvec, cg[2], xt, NF);
  zero_kernel<<<gBig, 256, 0, stream>>>(agg, N_NODES * NF);
  conv_edge_kernel<<<gE, 256, 0, stream>>>(xt, src, dstp, nedge, cmu[2], csg[2], agg);
  conv_finish_kernel<<<gN, 256, 0, stream>>>(agg, deg, dvec, crt[2], cbi[2], dvec, ybuf, NF, 1);

  // ---- decoder conv 3: no act, + d, straight into d_out ----
  conv_xt_kernel<<<gN, 256, 0, stream>>>(ybuf, cg[3], xt, NF);
  zero_kernel<<<gBig, 256, 0, stream>>>(agg, N_NODES * NF);
  conv_edge_kernel<<<gE, 256, 0, stream>>>(xt, src, dstp, nedge, cmu[3], csg[3], agg);
  conv_finish_kernel<<<gN, 256, 0, stream>>>(agg, deg, ybuf, crt[3], cbi[3], dvec,
                                             (float*)d_out, NF, 0);
}